// AFPConvolution_55121610277063
// MI455X (gfx1250) — compile-verified
//
#include <hip/hip_runtime.h>
#include <math.h>

typedef __attribute__((ext_vector_type(2))) float v2f;
typedef __attribute__((ext_vector_type(8))) float v8f;

#define DIM 64

// Fast transcendentals: straight-line code ending in hardware v_exp_f32
// (TRANS op, co-executes with VALU/WMMA; avoids libm's branchy expansion).
__device__ __forceinline__ float fast_exp(float x)     { return __expf(x); }
__device__ __forceinline__ float fast_sigmoid(float x) { return 1.0f / (1.0f + __expf(-x)); }
__device__ __forceinline__ float fast_tanh(float x)    { return 1.0f - 2.0f / (__expf(2.0f * x) + 1.0f); }

// ---------------------------------------------------------------------------
// Kernel 1: per-edge attention logit  s = leaky_relu([h_src | x_e] . w + b)
// ---------------------------------------------------------------------------
__global__ void edge_score_kernel(const float* __restrict__ node_emb,
                                  const float* __restrict__ nbr,
                                  const int*   __restrict__ bidx,
                                  const float* __restrict__ align_w,
                                  const float* __restrict__ align_b,
                                  float* __restrict__ scores,
                                  int n_edges) {
    int e = blockIdx.x * blockDim.x + threadIdx.x;
    if (e >= n_edges) return;
    int bi = bidx[e];
    const float4* nrow = (const float4*)(node_emb + (size_t)bi * DIM);
    const float4* erow = (const float4*)(nbr      + (size_t)e  * DIM);
    const float4* w0   = (const float4*)(align_w);
    const float4* w1   = (const float4*)(align_w + DIM);
    float s = align_b[0];
#pragma unroll
    for (int i = 0; i < DIM / 4; ++i) {
        float4 a = nrow[i], wa = w0[i];
        float4 b = erow[i], wb = w1[i];
        s += a.x * wa.x + a.y * wa.y + a.z * wa.z + a.w * wa.w;
        s += b.x * wb.x + b.y * wb.y + b.z * wb.z + b.w * wb.w;
    }
    scores[e] = (s > 0.0f) ? s : 0.01f * s;   // leaky_relu, slope 0.01
}

// ---------------------------------------------------------------------------
// Kernel 2: per-node segment softmax + attention-weighted pooling.
// batch_index is sorted -> contiguous edge ranges, found via binary search.
// One wave32 per node. pooled[n] = sum_e attn_e * nbr_e ; sum_attn in {0,1}.
// ---------------------------------------------------------------------------
__device__ __forceinline__ int lower_bound(const int* __restrict__ a, int n, int key) {
    int lo = 0, hi = n;
    while (lo < hi) {
        int mid = (lo + hi) >> 1;
        if (a[mid] < key) lo = mid + 1; else hi = mid;
    }
    return lo;
}

__global__ void node_pool_kernel(const float* __restrict__ nbr,
                                 const int*   __restrict__ bidx,
                                 const float* __restrict__ scores,
                                 float* __restrict__ pooled,
                                 float* __restrict__ sum_attn,
                                 int n_nodes, int n_edges) {
    int node = blockIdx.x;
    int lane = threadIdx.x;
    int start = lower_bound(bidx, n_edges, node);
    int end   = lower_bound(bidx, n_edges, node + 1);

    if (start >= end) {
        pooled[(size_t)node * DIM + lane]      = 0.0f;
        pooled[(size_t)node * DIM + lane + 32] = 0.0f;
        if (lane == 0) sum_attn[node] = 0.0f;
        return;
    }

    // segment max
    float mx = -3.4e38f;
    for (int e = start + lane; e < end; e += 32)
        mx = fmaxf(mx, scores[e]);
#pragma unroll
    for (int off = 16; off >= 1; off >>= 1)
        mx = fmaxf(mx, __shfl_xor(mx, off));

    // segment sum of exp
    float ssum = 0.0f;
    for (int e = start + lane; e < end; e += 32)
        ssum += fast_exp(scores[e] - mx);
#pragma unroll
    for (int off = 16; off >= 1; off >>= 1)
        ssum += __shfl_xor(ssum, off);

    float inv = 1.0f / fmaxf(ssum, 1e-16f);

    // weighted pooling: lane handles dims (lane, lane+32); coalesced rows
    float acc0 = 0.0f, acc1 = 0.0f;
    for (int e = start; e < end; ++e) {
        float w = fast_exp(scores[e] - mx) * inv;
        acc0 += w * nbr[(size_t)e * DIM + lane];
        acc1 += w * nbr[(size_t)e * DIM + lane + 32];
    }
    pooled[(size_t)node * DIM + lane]      = acc0;
    pooled[(size_t)node * DIM + lane + 32] = acc1;
    if (lane == 0) sum_attn[node] = 1.0f;
}

// ---------------------------------------------------------------------------
// Kernel 3: fused contexts GEMM (+ELU) and GRU readout, WMMA f32 16x16x4.
// One wave32 per 16-row tile. A-fragment layout (ISA 7.12.2, 32-bit A 16x4):
//   lane<16 : M=lane,    elements (K, K+1)
//   lane>=16: M=lane-16, elements (K+2, K+3)
// B[k][n] = W[n][k] -> B fragment is also a contiguous float2 from row n of W.
// C/D layout: VGPR r holds [M = r + 8*(lane>=16)][N = lane&15 (+n0)].
// ---------------------------------------------------------------------------
__device__ __forceinline__ v8f wmma_f32(v2f a, v2f b, v8f c) {
    return __builtin_amdgcn_wmma_f32_16x16x4_f32(false, a, false, b,
                                                 (short)0, c, false, false);
}

__global__ void readout_kernel(const float* __restrict__ node_emb,
                               const float* __restrict__ pooled,
                               const float* __restrict__ sum_attn,
                               const float* __restrict__ ctx_w,
                               const float* __restrict__ ctx_b,
                               const float* __restrict__ w_ih,
                               const float* __restrict__ w_hh,
                               const float* __restrict__ b_ih,
                               const float* __restrict__ b_hh,
                               float* __restrict__ out,
                               int n_nodes) {
    __shared__ float lds_ctx[16 * DIM];

    const int lane = threadIdx.x;
    const int half = lane >> 4;          // 0 | 1
    const int l15  = lane & 15;
    const int koff = half * 2;           // K offset within 4-chunk
    const int m0   = blockIdx.x * 16;

    const int arow = min(m0 + l15, n_nodes - 1);   // clamped A-fragment row

    // ---- preload A-fragments of pooled tile (16 chunks of K=4) ----
    v2f Apool[16];
#pragma unroll
    for (int kc = 0; kc < 16; ++kc) {
        const float* p = pooled + (size_t)arow * DIM + kc * 4 + koff;
        Apool[kc].x = p[0]; Apool[kc].y = p[1];
    }

    // ---- contexts = ELU(pooled @ ctx_w^T + sum_attn * ctx_b) -> LDS ----
#pragma unroll
    for (int t = 0; t < 4; ++t) {
        const int n0 = t * 16;
        v8f acc = {};
#pragma unroll
        for (int kc = 0; kc < 16; ++kc) {
            const float* bp = ctx_w + (size_t)(n0 + l15) * DIM + kc * 4 + koff;
            v2f b; b.x = bp[0]; b.y = bp[1];
            acc = wmma_f32(Apool[kc], b, acc);
        }
        const int col = n0 + l15;
        const float bias = ctx_b[col];
#pragma unroll
        for (int r = 0; r < 8; ++r) {
            const int row = r + half * 8;
            const float sa = sum_attn[min(m0 + row, n_nodes - 1)];
            float v = acc[r] + sa * bias;
            v = (v > 0.0f) ? v : (fast_exp(v) - 1.0f);       // ELU
            lds_ctx[row * DIM + col] = v;
        }
    }

    // ---- re-stripe contexts into A-fragment layout; load h fragments ----
    v2f Actx[16], Ah[16];
#pragma unroll
    for (int kc = 0; kc < 16; ++kc) {
        const float* cp = &lds_ctx[l15 * DIM + kc * 4 + koff];
        Actx[kc].x = cp[0]; Actx[kc].y = cp[1];
        const float* hp = node_emb + (size_t)arow * DIM + kc * 4 + koff;
        Ah[kc].x = hp[0]; Ah[kc].y = hp[1];
    }

    // ---- GRU: gi = ctx @ w_ih^T + b_ih ; gh = h @ w_hh^T + b_hh ----
#pragma unroll
    for (int t = 0; t < 4; ++t) {
        const int n0 = t * 16;
        v8f air = {}, aiz = {}, ain = {}, ahr = {}, ahz = {}, ahn = {};
#pragma unroll
        for (int kc = 0; kc < 16; ++kc) {
            const int kk = kc * 4 + koff;
            const size_t rn = (size_t)(n0 + l15) * DIM + kk;
            v2f br, bz, bn;
            br.x = w_ih[rn];                 br.y = w_ih[rn + 1];
            bz.x = w_ih[rn +  64 * DIM];     bz.y = w_ih[rn +  64 * DIM + 1];
            bn.x = w_ih[rn + 128 * DIM];     bn.y = w_ih[rn + 128 * DIM + 1];
            air = wmma_f32(Actx[kc], br, air);
            aiz = wmma_f32(Actx[kc], bz, aiz);
            ain = wmma_f32(Actx[kc], bn, ain);
            br.x = w_hh[rn];                 br.y = w_hh[rn + 1];
            bz.x = w_hh[rn +  64 * DIM];     bz.y = w_hh[rn +  64 * DIM + 1];
            bn.x = w_hh[rn + 128 * DIM];     bn.y = w_hh[rn + 128 * DIM + 1];
            ahr = wmma_f32(Ah[kc], br, ahr);
            ahz = wmma_f32(Ah[kc], bz, ahz);
            ahn = wmma_f32(Ah[kc], bn, ahn);
        }
        const int col = n0 + l15;
        const float bir = b_ih[col], biz = b_ih[64 + col], bin = b_ih[128 + col];
        const float bhr = b_hh[col], bhz = b_hh[64 + col], bhn = b_hh[128 + col];
#pragma unroll
        for (int r = 0; r < 8; ++r) {
            const int gm = m0 + r + half * 8;
            if (gm >= n_nodes) continue;
            const float h  = node_emb[(size_t)gm * DIM + col];
            const float rg = fast_sigmoid(air[r] + bir + ahr[r] + bhr);
            const float zg = fast_sigmoid(aiz[r] + biz + ahz[r] + bhz);
            const float g  = fast_tanh(ain[r] + bin + rg * (ahn[r] + bhn));
            const float hn = (1.0f - zg) * g + zg * h;
            out[(size_t)gm * DIM + col] = fmaxf(hn, 0.0f);
        }
    }
}

// ---------------------------------------------------------------------------
extern "C" void kernel_launch(void* const* d_in, const int* in_sizes, int n_in,
                              void* d_out, int out_size, void* d_ws, size_t ws_size,
                              hipStream_t stream) {
    const float* node_emb = (const float*)d_in[0];
    const float* nbr      = (const float*)d_in[1];
    const int*   bidx     = (const int*)  d_in[2];
    const float* align_w  = (const float*)d_in[3];
    const float* align_b  = (const float*)d_in[4];
    const float* ctx_w    = (const float*)d_in[5];
    const float* ctx_b    = (const float*)d_in[6];
    const float* w_ih     = (const float*)d_in[7];
    const float* w_hh     = (const float*)d_in[8];
    const float* b_ih     = (const float*)d_in[9];
    const float* b_hh     = (const float*)d_in[10];
    float* out = (float*)d_out;

    const int n_nodes = in_sizes[0] / DIM;
    const int n_edges = in_sizes[2];

    float* scores   = (float*)d_ws;                       // [E]
    float* pooled   = scores + n_edges;                   // [N*64]
    float* sum_attn = pooled + (size_t)n_nodes * DIM;     // [N]

    edge_score_kernel<<<(n_edges + 255) / 256, 256, 0, stream>>>(
        node_emb, nbr, bidx, align_w, align_b, scores, n_edges);

    node_pool_kernel<<<n_nodes, 32, 0, stream>>>(
        nbr, bidx, scores, pooled, sum_attn, n_nodes, n_edges);

    readout_kernel<<<(n_nodes + 15) / 16, 32, 0, stream>>>(
        node_emb, pooled, sum_attn, ctx_w, ctx_b,
        w_ih, w_hh, b_ih, b_hh, out, n_nodes);
}